// CayleyLinearApprox_24180665877092
// MI455X (gfx1250) — compile-verified
//
#include <hip/hip_runtime.h>

// CDNA5 / gfx1250, wave32. fp32 WMMA GEMM (V_WMMA_F32_16X16X4_F32) with
// double-buffered LDS fed by GLOBAL_LOAD_ASYNC_TO_LDS_B128 (ASYNCcnt path).

typedef __attribute__((ext_vector_type(2))) float v2f;
typedef __attribute__((ext_vector_type(8))) float v8f;

#define TM 128
#define TN 128
#define TK 32
#define LDA 36     // padded LDS row stride (floats) for A tile
#define LDB 136    // padded LDS row stride (floats) for B tile
#define A_TILE (TM * LDA)   // 4608 floats = 18 KB
#define B_TILE (TK * LDB)   // 4352 floats = 17 KB

// ---- CDNA5 async memory->LDS copy, 16 bytes per lane (VGLOBAL encoding, GV mode)
__device__ __forceinline__ void async_copy_b128(unsigned lds_byte_off,
                                                unsigned long long gaddr) {
  asm volatile("global_load_async_to_lds_b128 %0, %1, off"
               :
               : "v"(lds_byte_off), "v"(gaddr)
               : "memory");
}

__device__ __forceinline__ void wait_async0() {
  asm volatile("s_wait_asynccnt 0" ::: "memory");
}

// ---------------------------------------------------------------------------
// Expand angles -> M = I + A  (A skew-symmetric, upper-tri row-major packing)
// ---------------------------------------------------------------------------
__launch_bounds__(256)
__global__ void build_M_kernel(const float* __restrict__ angles,
                               float* __restrict__ Mout, int dim) {
  const int j = blockIdx.x * blockDim.x + threadIdx.x;
  const int i = blockIdx.y;
  float v;
  if (i == j) {
    v = 1.0f;
  } else if (i < j) {
    const long k = (long)i * (dim - 1) - ((long)i * (i - 1)) / 2 + (j - i - 1);
    v = angles[k];
  } else {
    const long k = (long)j * (dim - 1) - ((long)j * (j - 1)) / 2 + (i - j - 1);
    v = -angles[k];
  }
  Mout[(long)i * dim + j] = v;
}

// ---------------------------------------------------------------------------
// C[M,N] = A[M,K] @ B[K,N] (+ bias per column), fp32, WMMA 16x16x4.
// 256 threads = 8 waves; block tile 128x128; K-slab 32; double-buffered LDS.
// Wave grid 2x4 -> each wave computes 64x32 = 4x2 tiles of 16x16.
// ---------------------------------------------------------------------------
__launch_bounds__(256, 2)
__global__ void gemm_wmma_f32(const float* __restrict__ A,
                              const float* __restrict__ B,
                              float* __restrict__ C,
                              const float* __restrict__ bias,
                              int M, int N, int K) {
  __shared__ float As[2][A_TILE];   // 36 KB
  __shared__ float Bs[2][B_TILE];   // 34 KB

  const int  tid   = threadIdx.x;
  const int  lane  = tid & 31;
  const int  wave  = tid >> 5;
  const int  waveM = (wave >> 2) * 64;   // 0 or 64
  const int  waveN = (wave & 3) * 32;    // 0..96
  const long blockM = (long)blockIdx.y * TM;
  const long blockN = (long)blockIdx.x * TN;

  const int lm = lane & 15;          // M (A frag) / N (B frag) index
  const int lk = (lane >> 4) << 1;   // K base: 0 (lanes 0-15) or 2 (lanes 16-31)

  // ---- per-thread async-copy slots: 4 x b128 for A tile, 4 x b128 for B tile
  unsigned            a_lds[4], b_lds[4];
  unsigned long long  a_gbl[4], b_gbl[4];
#pragma unroll
  for (int i = 0; i < 4; ++i) {
    const int idx = tid + 256 * i;           // 0..1023
    const int ra = idx >> 3, ca = (idx & 7) << 2;     // A: 8 float4 per 32-float row
    a_lds[i] = (unsigned)(unsigned long long)(&As[0][ra * LDA + ca]);
    a_gbl[i] = (unsigned long long)(A + (blockM + ra) * (long)K + ca);
    const int rb = idx >> 5, cb = (idx & 31) << 2;    // B: 32 float4 per 128-float row
    b_lds[i] = (unsigned)(unsigned long long)(&Bs[0][rb * LDB + cb]);
    b_gbl[i] = (unsigned long long)(B + (long)rb * N + blockN + cb);
  }
  const unsigned a_buf = (unsigned)(A_TILE * sizeof(float));
  const unsigned b_buf = (unsigned)(B_TILE * sizeof(float));

  // ---- prologue: slab 0 -> buffer 0
#pragma unroll
  for (int i = 0; i < 4; ++i) {
    async_copy_b128(a_lds[i], a_gbl[i]);
    async_copy_b128(b_lds[i], b_gbl[i]);
  }

  v8f acc[4][2];
  const v8f vzero = {0.f, 0.f, 0.f, 0.f, 0.f, 0.f, 0.f, 0.f};
#pragma unroll
  for (int mt = 0; mt < 4; ++mt)
#pragma unroll
    for (int nt = 0; nt < 2; ++nt)
      acc[mt][nt] = vzero;

  const int nslab = K / TK;
  for (int kb = 0; kb < nslab; ++kb) {
    const int cur = kb & 1;

    wait_async0();      // this wave's copies into buf 'cur' have landed
    __syncthreads();    // ...and every other wave's too; prev slab reads done

    if (kb + 1 < nslab) {
      const unsigned nxt = (unsigned)(cur ^ 1);
      const unsigned long long aoff =
          (unsigned long long)(kb + 1) * (TK * sizeof(float));
      const unsigned long long boff =
          (unsigned long long)(kb + 1) * ((unsigned long long)TK * N * sizeof(float));
#pragma unroll
      for (int i = 0; i < 4; ++i) {
        async_copy_b128(a_lds[i] + nxt * a_buf, a_gbl[i] + aoff);
        async_copy_b128(b_lds[i] + nxt * b_buf, b_gbl[i] + boff);
      }
    }

    const float* as = &As[cur][0];
    const float* bs = &Bs[cur][0];

    // ---- compute: 8 k-steps of 4, 8 WMMAs each
#pragma unroll
    for (int k4 = 0; k4 < TK; k4 += 4) {
      v2f af[4];
      v2f bf[2];
#pragma unroll
      for (int mt = 0; mt < 4; ++mt) {
        // A 16x4 frag: lane = m + 16*(k/2), reg = k%2 -> contiguous pair (b64)
        const float* p = &as[(waveM + mt * 16 + lm) * LDA + k4 + lk];
        af[mt] = *(const v2f*)p;
      }
#pragma unroll
      for (int nt = 0; nt < 2; ++nt) {
        // B 4x16 frag: reg r = B[k4+lk+r][n], n = lane%16
        const float* p = &bs[(k4 + lk) * LDB + waveN + nt * 16 + lm];
        v2f b;
        b[0] = p[0];
        b[1] = p[LDB];
        bf[nt] = b;
      }
#pragma unroll
      for (int mt = 0; mt < 4; ++mt)
#pragma unroll
        for (int nt = 0; nt < 2; ++nt)
          acc[mt][nt] = __builtin_amdgcn_wmma_f32_16x16x4_f32(
              false, af[mt], false, bf[nt], (short)0, acc[mt][nt], false, false);
    }
  }

  // ---- epilogue: C/D layout = VGPR v holds M = v + 8*(lane>=16), N = lane%16
  const int mh = (lane >> 4) << 3;   // 0 or 8
#pragma unroll
  for (int nt = 0; nt < 2; ++nt) {
    const long n  = blockN + waveN + nt * 16 + lm;
    const float bv = bias ? bias[n] : 0.0f;
#pragma unroll
    for (int mt = 0; mt < 4; ++mt) {
      const long mbase = blockM + waveM + mt * 16 + mh;
#pragma unroll
      for (int v = 0; v < 8; ++v) {
        C[(mbase + v) * (long)N + n] = acc[mt][nt][v] + bv;
      }
    }
  }
}

// ---------------------------------------------------------------------------
// Launch: build M = I + A ; W = M @ M ; out = x @ W + bias
// ---------------------------------------------------------------------------
extern "C" void kernel_launch(void* const* d_in, const int* in_sizes, int n_in,
                              void* d_out, int out_size, void* d_ws, size_t ws_size,
                              hipStream_t stream) {
  const float* x      = (const float*)d_in[0];   // [4, 4096, 2048] -> 16384 x 2048
  const float* angles = (const float*)d_in[1];   // dim*(dim-1)/2
  const float* bias   = (const float*)d_in[2];   // [2048]
  float* out = (float*)d_out;

  const int dim  = 2048;
  const int rows = 4 * 4096;  // 16384

  float* Mws = (float*)d_ws;                     // 16 MB
  float* Wws = Mws + (size_t)dim * dim;          // 16 MB

  build_M_kernel<<<dim3(dim / 256, dim), 256, 0, stream>>>(angles, Mws, dim);

  // W = (I+A) @ (I+A)
  gemm_wmma_f32<<<dim3(dim / TN, dim / TM), 256, 0, stream>>>(
      Mws, Mws, Wws, nullptr, dim, dim, dim);

  // out = x @ W + bias
  gemm_wmma_f32<<<dim3(dim / TN, rows / TM), 256, 0, stream>>>(
      x, Wws, out, bias, rows, dim, dim);
}